// HandsToMask_36876589204231
// MI455X (gfx1250) — compile-verified
//
#include <hip/hip_runtime.h>
#include <stdint.h>

// HandsToMask, MI455X (gfx1250).
// Pure bandwidth kernel: 402.7 MB of write-once output vs ~0 FLOPs -> the
// roofline is the HBM store stream (~17.3 us @ 23.3 TB/s). No WMMA mapping
// makes sense (zero matrix math); the CDNA5 features that do pay are the
// Tensor Data Mover (hands tile -> LDS once per workgroup), LDS broadcast of
// the 64-way shared hand row, and non-temporal b128 streaming stores.

typedef unsigned int v4u __attribute__((ext_vector_type(4)));
typedef int          v8i __attribute__((ext_vector_type(8)));
typedef int          v4i __attribute__((ext_vector_type(4)));
typedef float        v4f __attribute__((ext_vector_type(4)));

#define H_CARDS            14
#define NUM_CARD           512
#define OUT_W              (NUM_CARD * 3)          /* 1536 floats per row   */
#define ROWS_PER_BLOCK     4
#define THREADS_PER_ROW    64
#define BLOCK_SIZE         (ROWS_PER_BLOCK * THREADS_PER_ROW)  /* 256 = 8 waves */
#define GROUPS_PER_THREAD  8                        /* card groups per thread */
#define FLOATS_PER_THREAD  (GROUPS_PER_THREAD * 3)  /* 24 floats = 6 x b128  */
#define TILE_I32           (ROWS_PER_BLOCK * H_CARDS)  /* 56 ints = 224 B    */

__global__ __launch_bounds__(BLOCK_SIZE)
void hands_to_mask_kernel(const int* __restrict__ hands,
                          float* __restrict__ out) {
  __shared__ __align__(16) int sh[TILE_I32];

  // ---- TDM: DMA this block's 4 hand rows (56 contiguous i32) into LDS ----
  if (threadIdx.x == 0) {
    unsigned long long ga =
        (unsigned long long)(uintptr_t)hands +
        (unsigned long long)blockIdx.x * (unsigned long long)(TILE_I32 * 4);
    // Low 32 bits of the generic pointer to LDS == LDS byte offset.
    unsigned lds_off = (unsigned)(uintptr_t)(&sh[0]);

    // D# group 0: count=1 | lds_addr | global_addr[56:0] | type=2
    v4u g0;
    g0.x = 1u;
    g0.y = lds_off;
    g0.z = (unsigned)(ga & 0xffffffffull);
    g0.w = (unsigned)((ga >> 32) & 0x01ffffffull) | (2u << 30);

    // D# group 1: data_size=4B; tensor_dim0=56; tensor_dim1=1;
    //             tile_dim0=56 (1-D tile, tile_dim1=0); dim0_stride=56.
    v8i g1;
    g1[0] = (2 << 16);          // workgroup_mask=0, data_size=2 (4 bytes)
    g1[1] = (TILE_I32 << 16);   // tensor_dim0[15:0] at bits 63:48
    g1[2] = (1 << 16);          // tensor_dim0[31:16]=0, tensor_dim1[15:0]=1
    g1[3] = (TILE_I32 << 16);   // tensor_dim1[31:16]=0, tile_dim0=56
    g1[4] = 0;                  // tile_dim1=0 (unused), tile_dim2=0
    g1[5] = TILE_I32;           // tensor_dim0_stride[31:0]
    g1[6] = 0;
    g1[7] = 0;

    v4i gz4 = {0, 0, 0, 0};               // groups 2/3 unused (<=2D tensor)
    v8i gz8 = {0, 0, 0, 0, 0, 0, 0, 0};   // trailing descriptor group unused
    // 6-arg form (clang-23 / therock-10.0 headers):
    // (v4u g0, v8i g1, v4i g2, v4i g3, v8i g4, i32 cpol)
    __builtin_amdgcn_tensor_load_to_lds(g0, g1, gz4, gz4, gz8, 0);
    __builtin_amdgcn_s_wait_tensorcnt(0);   // s_wait_tensorcnt 0
  }
  __syncthreads();  // publish LDS tile to all 8 waves

  // ---- Gather: each thread owns 8 card groups = 24 output floats ----
  const int t          = threadIdx.x;
  const int row_in_blk = t >> 6;                    // 0..3
  const int lane       = t & (THREADS_PER_ROW - 1); // 0..63
  const int G          = lane * GROUPS_PER_THREAD;  // first card group, 0..511

  unsigned mask = 0u;
#pragma unroll
  for (int h = 0; h < H_CARDS; ++h) {
    int v = sh[row_in_blk * H_CARDS + h];           // ds_load (64-way shared)
    unsigned rel = (unsigned)(v - 1 - G);           // empty (v==0) -> huge
    if (rel < (unsigned)GROUPS_PER_THREAD)
      mask |= 1u << rel;
  }

  float of[FLOATS_PER_THREAD];
#pragma unroll
  for (int g = 0; g < GROUPS_PER_THREAD; ++g) {
    float v = ((mask >> g) & 1u) ? 0.0f : -100.0f;
    of[3 * g + 0] = v;
    of[3 * g + 1] = v;
    of[3 * g + 2] = v;
  }

  const size_t row = (size_t)blockIdx.x * ROWS_PER_BLOCK + row_in_blk;
  float* outp = out + row * OUT_W + (size_t)lane * FLOATS_PER_THREAD;
#pragma unroll
  for (int i = 0; i < FLOATS_PER_THREAD / 4; ++i) {
    v4f val = { of[4 * i + 0], of[4 * i + 1], of[4 * i + 2], of[4 * i + 3] };
    __builtin_nontemporal_store(val, (v4f*)outp + i);  // global_store_b128 nt
  }
}

extern "C" void kernel_launch(void* const* d_in, const int* in_sizes, int n_in,
                              void* d_out, int out_size, void* d_ws, size_t ws_size,
                              hipStream_t stream) {
  const int* hands = (const int*)d_in[0];   // [B, 14] int32
  float* out = (float*)d_out;               // [B, 1536] float32
  (void)n_in; (void)d_ws; (void)ws_size; (void)out_size;

  const int B = in_sizes[0] / H_CARDS;      // 65536
  const int grid = (B + ROWS_PER_BLOCK - 1) / ROWS_PER_BLOCK;  // 16384
  hands_to_mask_kernel<<<grid, BLOCK_SIZE, 0, stream>>>(hands, out);
}